// AttentionGuidedMaskStrategy_33715493274243
// MI455X (gfx1250) — compile-verified
//
#include <hip/hip_runtime.h>
#include <hip/hip_bf16.h>

typedef __attribute__((ext_vector_type(2))) float v2f;
typedef __attribute__((ext_vector_type(8))) float v8f;

#define B_ 64
#define L_ 512
#define E_ 256
#define NCHUNK (L_ / 16)   // 32 key chunks of width 16

// ---------------------------------------------------------------------------
// Kernel 1: masked column sums of attn via V_WMMA_F32_16X16X4_F32.
// One wave (32 threads) per (side, batch, 16-key chunk).
//   D = A(16x4) * B(4x16) + C accumulated over q in steps of 4.
// Padding is a suffix (pad = arange >= len), so the masked column sum equals
// the sum over q in [0, len):
//   - main loop: A = all-ones constant, SCALAR-uniform trip count
//     (readfirstlane) so the loop is a uniform s_cmp loop -> EXEC stays
//     all-ones at every WMMA, and the loop can unroll/pipeline loads
//   - one boundary WMMA (uniform s_cbranch) handles len % 4
// Operand layout (matches documented 16x4 f32 A layout; B mirrored):
//   lanes 0..15  : K = 0,1 in the two operand registers
//   lanes 16..31 : K = 2,3
// ---------------------------------------------------------------------------
__global__ __launch_bounds__(32) void colsum_wmma_kernel(
    const float* __restrict__ attn_a, const float* __restrict__ attn_b,
    const unsigned char* __restrict__ a_pad, const unsigned char* __restrict__ b_pad,
    float* __restrict__ keysum /* [2][B][L] */)
{
    int id    = blockIdx.x;                 // side*B*NCHUNK + batch*NCHUNK + chunk
    int chunk = id & (NCHUNK - 1);
    int batch = (id >> 5) & (B_ - 1);
    int side  = id >> 11;

    const float* attn = side ? attn_b : attn_a;
    const unsigned char* qpad = (side ? b_pad : a_pad) + (size_t)batch * L_;
    const float* base = attn + (size_t)batch * L_ * L_ + chunk * 16;

    int lane = threadIdx.x & 31;
    int n    = lane & 15;       // key within chunk
    int kg   = lane >> 4;       // K group: 0 -> K=0,1 ; 1 -> K=2,3

    // ---- number of non-padded queries: pad bytes are 0/1, so popcount works.
    const uint4* p4 = (const uint4*)qpad;           // 512 B = 32 lanes * 16 B
    uint4 pw = p4[lane];
    int npad = __popc(pw.x) + __popc(pw.y) + __popc(pw.z) + __popc(pw.w);
    for (int off = 16; off; off >>= 1)
        npad += __shfl_xor(npad, off, 32);
    // Force into an SGPR: every lane holds the full sum, and a scalar trip
    // count keeps EXEC all-ones (WMMA requirement) + enables unrolling.
    int len  = __builtin_amdgcn_readfirstlane(L_ - npad);
    int lenf = len & ~3;                            // full K=4 groups

    v2f ones;
    ones[0] = 1.0f; ones[1] = 1.0f;

    const float* p = base + (size_t)(kg * 2) * L_ + n;  // row q0 + kg*2, col n
    v8f c = {};
#pragma unroll 4
    for (int q0 = 0; q0 < lenf; q0 += 4) {
        v2f bt;
        bt[0] = p[0];
        bt[1] = p[L_];
        c = __builtin_amdgcn_wmma_f32_16x16x4_f32(
                false, ones, false, bt, (short)0, c, false, false);
        p += 4 * L_;
    }
    int rem = len - lenf;                           // 0..3, scalar branch
    if (rem) {
        // rows lenf .. lenf+3 are in-bounds (lenf <= 508 whenever rem > 0)
        v2f at, bt;
        at[0] = (kg * 2 + 0 < rem) ? 1.0f : 0.0f;
        at[1] = (kg * 2 + 1 < rem) ? 1.0f : 0.0f;
        bt[0] = p[0];
        bt[1] = p[L_];
        c = __builtin_amdgcn_wmma_f32_16x16x4_f32(
                false, at, false, bt, (short)0, c, false, false);
    }
    // D row 0 lives in VGPR0 of lanes 0..15 (N = lane)
    if (lane < 16)
        keysum[((size_t)side * B_ + batch) * L_ + chunk * 16 + n] = c[0];
}

// ---------------------------------------------------------------------------
// Kernel 2: per-batch stable ranking -> 0/1 mask.
// mask[j] = (rank_of_j_among_ascending(vals, zeros->+inf) < int(0.3*#nonpadded))
//           && vals[j] != 0
// One 512-thread block per (side, batch).
// ---------------------------------------------------------------------------
__global__ __launch_bounds__(L_) void rank_mask_kernel(
    const float* __restrict__ keysum,
    const unsigned char* __restrict__ a_pad, const unsigned char* __restrict__ b_pad,
    float* __restrict__ maskbuf /* [2][B][L] */)
{
    __shared__ float vals[L_];
    __shared__ int cnt;

    int side  = blockIdx.x >> 6;
    int batch = blockIdx.x & (B_ - 1);
    // side 0: keys of attn_a (Lb) -> k_pad = b_padding_mask
    // side 1: keys of attn_b (La) -> k_pad = a_padding_mask
    const unsigned char* kpad = (side ? a_pad : b_pad) + (size_t)batch * L_;

    int j = threadIdx.x;
    float v  = keysum[((size_t)side * B_ + batch) * L_ + j];
    bool nnz = (v != 0.0f);
    float vp = nnz ? v : __builtin_inff();

    if (j == 0) cnt = 0;
    vals[j] = vp;
    __syncthreads();
    if (!kpad[j]) atomicAdd(&cnt, 1);
    __syncthreads();

    int kth = (int)(0.3f * (float)cnt);   // f32 multiply + truncation, as reference

    int rank = 0;
    for (int i = 0; i < L_; ++i) {
        float vi = vals[i];
        rank += (int)((vi < vp) | ((vi == vp) & (i < j)));   // stable ascending
    }
    maskbuf[((size_t)side * B_ + batch) * L_ + j] =
        (nnz && (rank < kth)) ? 1.0f : 0.0f;
}

// ---------------------------------------------------------------------------
// Kernel 3: streaming blend, float4 vectorized.
// out = (1-m)*embed + m*mask_embedding
// Output layout: [masked_embed_b (side0) | masked_embed_a (side1)], each [B,L,E].
// ---------------------------------------------------------------------------
__global__ __launch_bounds__(256) void blend_kernel(
    const float* __restrict__ embed_a, const float* __restrict__ embed_b,
    const float* __restrict__ mask_emb, const float* __restrict__ maskbuf,
    float* __restrict__ out)
{
    size_t gid = (size_t)blockIdx.x * blockDim.x + threadIdx.x;  // over 2*B*L*(E/4)
    int e4   = (int)(gid & (E_ / 4 - 1));
    size_t t = gid >> 6;                 // token index: side*B*L + batch*L + j
    int side = (int)(t >> 15);           // B*L = 32768 = 2^15

    float m  = maskbuf[t];
    float om = 1.0f - m;

    const float4* emb = (const float4*)(side ? embed_a : embed_b);
    size_t pos = t - ((size_t)side << 15);

    float4 x  = emb[pos * (E_ / 4) + e4];
    float4 me = ((const float4*)mask_emb)[e4];
    float4 r;
    r.x = om * x.x + m * me.x;
    r.y = om * x.y + m * me.y;
    r.z = om * x.z + m * me.z;
    r.w = om * x.w + m * me.w;
    ((float4*)out)[gid] = r;
}

extern "C" void kernel_launch(void* const* d_in, const int* in_sizes, int n_in,
                              void* d_out, int out_size, void* d_ws, size_t ws_size,
                              hipStream_t stream) {
    (void)in_sizes; (void)n_in; (void)out_size; (void)ws_size;

    const float*         attn_a   = (const float*)d_in[0];
    const float*         attn_b   = (const float*)d_in[1];
    const float*         embed_a  = (const float*)d_in[2];
    const float*         embed_b  = (const float*)d_in[3];
    const float*         mask_emb = (const float*)d_in[4];
    const unsigned char* a_pad    = (const unsigned char*)d_in[5];
    const unsigned char* b_pad    = (const unsigned char*)d_in[6];
    float* out = (float*)d_out;

    float* keysum  = (float*)d_ws;                 // [2][B][L] floats
    float* maskbuf = keysum + (size_t)2 * B_ * L_; // [2][B][L] floats

    colsum_wmma_kernel<<<2 * B_ * NCHUNK, 32, 0, stream>>>(
        attn_a, attn_b, a_pad, b_pad, keysum);

    rank_mask_kernel<<<2 * B_, L_, 0, stream>>>(
        keysum, a_pad, b_pad, maskbuf);

    size_t nvec = (size_t)2 * B_ * L_ * (E_ / 4);
    blend_kernel<<<(unsigned)(nvec / 256), 256, 0, stream>>>(
        embed_a, embed_b, mask_emb, maskbuf, out);
}